// Attention_50637664420407
// MI455X (gfx1250) — compile-verified
//
#include <hip/hip_runtime.h>
#include <math.h>

#define SEQ    2048
#define BATCH  2
#define DIM    1024
#define HEADS  16
#define DHEAD  64
#define INNER  1024
#define ROWS   (BATCH * SEQ) /* 4096 */

typedef float v2f __attribute__((ext_vector_type(2)));
typedef float v8f __attribute__((ext_vector_type(8)));

__device__ __forceinline__ v8f wmma_f32(v2f a, v2f b, v8f c) {
  // (neg_a, A, neg_b, B, c_mod, C, reuse_a, reuse_b)
  return __builtin_amdgcn_wmma_f32_16x16x4_f32(false, a, false, b, (short)0, c,
                                               false, false);
}

// ---------------------------------------------------------------- LayerNorm
__global__ __launch_bounds__(256) void ln_kernel(const float* __restrict__ x,
                                                 const float* __restrict__ g,
                                                 const float* __restrict__ be,
                                                 float* __restrict__ xn) {
  const int row = blockIdx.x;
  const int tid = threadIdx.x;           // 256 threads, 4 floats each
  const float4* xr = (const float4*)(x + (size_t)row * DIM);
  float4 v = xr[tid];

  __shared__ float red[256];
  red[tid] = v.x + v.y + v.z + v.w;
  __syncthreads();
  for (int off = 128; off > 0; off >>= 1) {
    if (tid < off) red[tid] += red[tid + off];
    __syncthreads();
  }
  float mu = red[0] * (1.0f / DIM);
  __syncthreads();

  float dx = v.x - mu, dy = v.y - mu, dz = v.z - mu, dw = v.w - mu;
  red[tid] = dx * dx + dy * dy + dz * dz + dw * dw;
  __syncthreads();
  for (int off = 128; off > 0; off >>= 1) {
    if (tid < off) red[tid] += red[tid + off];
    __syncthreads();
  }
  float r = rsqrtf(red[0] * (1.0f / DIM) + 1e-5f);

  float4 gg = ((const float4*)g)[tid];
  float4 bb = ((const float4*)be)[tid];
  float4 o;
  o.x = dx * r * gg.x + bb.x;
  o.y = dy * r * gg.y + bb.y;
  o.z = dz * r * gg.z + bb.z;
  o.w = dw * r * gg.w + bb.w;
  ((float4*)(xn + (size_t)row * DIM))[tid] = o;
}

// ------------------------------------------------------ QKV GEMM (fp32 WMMA)
// C[4096,3072] = xn[4096,1024] @ W_qkv[1024,3072].
// One wave computes a 16(M) x 64(N) tile: the A fragment is reused by 4
// WMMAs per k-step. Epilogue scatters into Q/K/V buffers laid out [b,h,n,d].
__global__ __launch_bounds__(32) void qkv_gemm(const float* __restrict__ A,
                                               const float* __restrict__ W,
                                               float* __restrict__ Qb,
                                               float* __restrict__ Kb,
                                               float* __restrict__ Vb) {
  const int NCOL  = 3 * INNER;               // 3072
  const int tileN = blockIdx.x;              // 0..47 (64 cols each)
  const int tileM = blockIdx.y;              // 0..255
  const int lane  = threadIdx.x;
  const int l15   = lane & 15;
  const int half  = lane >> 4;
  const int khalf = half << 1;               // 0 or 2

  const float* Arow = A + (size_t)(tileM * 16 + l15) * DIM;
  const int col0 = tileN * 64 + l15;

  v8f acc[4] = {};
#pragma unroll 4
  for (int k = 0; k < DIM; k += 4) {
    v2f a = *(const v2f*)(Arow + k + khalf);
    const float* Wr0 = W + (size_t)(k + khalf) * NCOL + col0;
    const float* Wr1 = Wr0 + NCOL;
#pragma unroll
    for (int t = 0; t < 4; ++t) {
      v2f b;
      b.x = Wr0[t * 16];
      b.y = Wr1[t * 16];
      acc[t] = wmma_f32(a, b, acc[t]);
    }
  }

  // scatter: col -> (which, head, dim)
#pragma unroll
  for (int t = 0; t < 4; ++t) {
    const int col    = col0 + t * 16;
    const int which  = col / INNER;
    const int within = col % INNER;
    const int h      = within / DHEAD;
    const int d      = within % DHEAD;
    float* dst = (which == 0) ? Qb : ((which == 1) ? Kb : Vb);
#pragma unroll
    for (int j = 0; j < 8; ++j) {
      int r  = tileM * 16 + (half << 3) + j; // global row in [0,4096)
      int bb = r / SEQ;
      int n  = r % SEQ;
      dst[(((size_t)(bb * HEADS + h) * SEQ + n) * DHEAD) + d] = acc[t][j];
    }
  }
}

// ------------------------------------------------------------------- RoPE
__global__ __launch_bounds__(256) void rope_kernel(float* __restrict__ Qb,
                                                   float* __restrict__ Kb) {
  size_t idx = (size_t)blockIdx.x * blockDim.x + threadIdx.x; // b*h*n*32
  int j    = (int)(idx & 31);              // rotation pair index
  size_t t = idx >> 5;                     // (b*h)*SEQ + n
  int n    = (int)(t % SEQ);
  size_t base = t * DHEAD;

  float inv = powf(10000.0f, -(float)j * (1.0f / 32.0f));
  float ang = (float)n * inv;
  float c = cosf(ang), s = sinf(ang);

  float q0 = Qb[base + j], q1 = Qb[base + j + 32];
  Qb[base + j]      = q0 * c - q1 * s;
  Qb[base + j + 32] = q1 * c + q0 * s;
  float k0 = Kb[base + j], k1 = Kb[base + j + 32];
  Kb[base + j]      = k0 * c - k1 * s;
  Kb[base + j + 32] = k1 * c + k0 * s;
}

// --------------------------------------------- Flash attention (fp32 WMMA)
// One wave handles one (b,h) x 16-row query tile. Online softmax; LDS used
// to move P from the C-matrix layout into the A-matrix layout.
__global__ __launch_bounds__(32) void attn_kernel(const float* __restrict__ Qb,
                                                  const float* __restrict__ Kb,
                                                  const float* __restrict__ Vb,
                                                  float* __restrict__ O) {
  const int qt   = blockIdx.x;             // 0..127
  const int bh   = blockIdx.y;             // 0..31
  const int h    = bh % HEADS;
  const int b    = bh / HEADS;
  const int lane = threadIdx.x;
  const int l15  = lane & 15;
  const int half = lane >> 4;
  const int khalf = half << 1;

  const float* Qh = Qb + (size_t)bh * SEQ * DHEAD;
  const float* Kh = Kb + (size_t)bh * SEQ * DHEAD;
  const float* Vh = Vb + (size_t)bh * SEQ * DHEAD;

  // Preload this lane's slice of the Q tile (A-operand layout).
  v2f qreg[16];
  const float* qrow = Qh + (size_t)(qt * 16 + l15) * DHEAD;
#pragma unroll
  for (int t = 0; t < 16; ++t) qreg[t] = *(const v2f*)(qrow + 4 * t + khalf);

  v8f acc0 = {}, acc1 = {}, acc2 = {}, acc3 = {};
  float mrow[8], lrow[8];
#pragma unroll
  for (int j = 0; j < 8; ++j) { mrow[j] = -INFINITY; lrow[j] = 0.0f; }

  __shared__ float P[16 * 17];             // padded to dodge bank conflicts
  const float scale = 0.125f;              // 64^-0.5

  for (int jt = 0; jt <= qt; ++jt) {
    // ---- S = Q * K^T (16x16, K-dim = 64) ----
    v8f s = {};
    const float* krow = Kh + (size_t)(jt * 16 + l15) * DHEAD;
#pragma unroll
    for (int t = 0; t < 16; ++t) {
      v2f bb = *(const v2f*)(krow + 4 * t + khalf);
      s = wmma_f32(qreg[t], bb, s);
    }

    // ---- online softmax (rows owned per C-layout) ----
    const int colg = jt * 16 + l15;
    float alpha[8];
#pragma unroll
    for (int j = 0; j < 8; ++j) {
      int rowg = qt * 16 + half * 8 + j;
      float sj = s[j] * scale;
      if (colg > rowg) sj = -INFINITY;     // causal mask
      float mx = sj;
      mx = fmaxf(mx, __shfl_xor(mx, 1, 16));
      mx = fmaxf(mx, __shfl_xor(mx, 2, 16));
      mx = fmaxf(mx, __shfl_xor(mx, 4, 16));
      mx = fmaxf(mx, __shfl_xor(mx, 8, 16));
      float mnew = fmaxf(mrow[j], mx);
      float p  = __expf(sj - mnew);
      float ps = p;
      ps += __shfl_xor(ps, 1, 16);
      ps += __shfl_xor(ps, 2, 16);
      ps += __shfl_xor(ps, 4, 16);
      ps += __shfl_xor(ps, 8, 16);
      float al = __expf(mrow[j] - mnew);
      lrow[j] = lrow[j] * al + ps;
      mrow[j] = mnew;
      alpha[j] = al;
      P[(half * 8 + j) * 17 + l15] = p;    // C-layout -> LDS
    }
#pragma unroll
    for (int j = 0; j < 8; ++j) {
      acc0[j] *= alpha[j]; acc1[j] *= alpha[j];
      acc2[j] *= alpha[j]; acc3[j] *= alpha[j];
    }
    __syncthreads();                       // single wave: orders DS ops

    // ---- O += P * V (16x64, K-dim = 16 keys) ----
#pragma unroll
    for (int ks = 0; ks < 4; ++ks) {
      v2f a;
      a.x = P[l15 * 17 + 4 * ks + khalf];
      a.y = P[l15 * 17 + 4 * ks + khalf + 1];
      const float* v0 = Vh + (size_t)(jt * 16 + 4 * ks + khalf) * DHEAD;
      const float* v1 = v0 + DHEAD;
      v2f b0; b0.x = v0[l15];      b0.y = v1[l15];
      acc0 = wmma_f32(a, b0, acc0);
      v2f b1; b1.x = v0[16 + l15]; b1.y = v1[16 + l15];
      acc1 = wmma_f32(a, b1, acc1);
      v2f b2; b2.x = v0[32 + l15]; b2.y = v1[32 + l15];
      acc2 = wmma_f32(a, b2, acc2);
      v2f b3; b3.x = v0[48 + l15]; b3.y = v1[48 + l15];
      acc3 = wmma_f32(a, b3, acc3);
    }
    __syncthreads();
  }

  // ---- normalize + store to [b, n, h*64+d] ----
#pragma unroll
  for (int j = 0; j < 8; ++j) {
    float invl = 1.0f / lrow[j];
    int rowg = qt * 16 + half * 8 + j;
    size_t off = ((size_t)(b * SEQ + rowg)) * INNER + h * DHEAD;
    O[off +      l15] = acc0[j] * invl;
    O[off + 16 + l15] = acc1[j] * invl;
    O[off + 32 + l15] = acc2[j] * invl;
    O[off + 48 + l15] = acc3[j] * invl;
  }
}

// --------------------------------------------- Output projection (fp32 WMMA)
// One wave computes a 16(M) x 64(N) tile with shared A fragments.
__global__ __launch_bounds__(32) void out_gemm(const float* __restrict__ A,
                                               const float* __restrict__ W,
                                               const float* __restrict__ bias,
                                               float* __restrict__ C) {
  const int tileN = blockIdx.x;            // 0..15 (64 cols each)
  const int tileM = blockIdx.y;            // 0..255
  const int lane  = threadIdx.x;
  const int l15   = lane & 15;
  const int half  = lane >> 4;
  const int khalf = half << 1;

  const float* Arow = A + (size_t)(tileM * 16 + l15) * INNER;
  const int col0 = tileN * 64 + l15;

  v8f acc[4] = {};
#pragma unroll 4
  for (int k = 0; k < INNER; k += 4) {
    v2f a = *(const v2f*)(Arow + k + khalf);
    const float* Wr0 = W + (size_t)(k + khalf) * DIM + col0;
    const float* Wr1 = Wr0 + DIM;
#pragma unroll
    for (int t = 0; t < 4; ++t) {
      v2f b;
      b.x = Wr0[t * 16];
      b.y = Wr1[t * 16];
      acc[t] = wmma_f32(a, b, acc[t]);
    }
  }

#pragma unroll
  for (int t = 0; t < 4; ++t) {
    const int col = col0 + t * 16;
    float bc = bias[col];
#pragma unroll
    for (int j = 0; j < 8; ++j) {
      int r = tileM * 16 + (half << 3) + j;
      C[(size_t)r * DIM + col] = acc[t][j] + bc;
    }
  }
}

// --------------------------------------------------------------------------
extern "C" void kernel_launch(void* const* d_in, const int* in_sizes, int n_in,
                              void* d_out, int out_size, void* d_ws,
                              size_t ws_size, hipStream_t stream) {
  const float* x     = (const float*)d_in[0];
  const float* W_qkv = (const float*)d_in[1];
  const float* W_out = (const float*)d_in[2];
  const float* b_out = (const float*)d_in[3];
  const float* ln_g  = (const float*)d_in[4];
  const float* ln_b  = (const float*)d_in[5];

  float* ws = (float*)d_ws;
  const size_t qkv_elems = (size_t)BATCH * HEADS * SEQ * DHEAD; // 4M floats
  float* xn = ws;                                   // [4096,1024]
  float* Qb = ws + (size_t)ROWS * DIM;
  float* Kb = Qb + qkv_elems;
  float* Vb = Kb + qkv_elems;
  float* O  = xn;                                   // reuse after QKV GEMM

  ln_kernel<<<ROWS, 256, 0, stream>>>(x, ln_g, ln_b, xn);

  dim3 g1(3 * INNER / 64, ROWS / 16);               // 48 x 256
  qkv_gemm<<<g1, 32, 0, stream>>>(xn, W_qkv, Qb, Kb, Vb);

  rope_kernel<<<(BATCH * HEADS * SEQ * 32) / 256, 256, 0, stream>>>(Qb, Kb);

  dim3 g2(SEQ / 16, BATCH * HEADS);                 // 128 x 32
  attn_kernel<<<g2, 32, 0, stream>>>(Qb, Kb, Vb, O);

  dim3 g3(DIM / 64, ROWS / 16);                     // 16 x 256
  out_gemm<<<g3, 32, 0, stream>>>(O, W_out, b_out, (float*)d_out);
}